// LinearAttentionLayer_40278203302000
// MI455X (gfx1250) — compile-verified
//
#include <hip/hip_runtime.h>
#include <hip/hip_bf16.h>
#include <math.h>

// ---------------------------------------------------------------------------
// LinearAttentionLayer for MI455X (gfx1250), fp32 end-to-end via
// V_WMMA_F32_16X16X4_F32.  B=2, T=2048, D=256.
// ---------------------------------------------------------------------------

#define BB 2
#define TT 2048
#define DD 256
#define BT (BB * TT)     // 4096 rows
#define D3 (3 * DD)      // 768
#define HH (4 * DD)      // 1024

typedef float v8f __attribute__((ext_vector_type(8)));
typedef float v2f __attribute__((ext_vector_type(2)));

__device__ __forceinline__ v8f wmma4(v2f a, v2f b, v8f c) {
  // D = A(16x4,f32) * B(4x16,f32) + C(16x16,f32)
  return __builtin_amdgcn_wmma_f32_16x16x4_f32(
      /*neg_a=*/false, a, /*neg_b=*/false, b,
      /*c_mod=*/(short)0, c, /*reuse_a=*/false, /*reuse_b=*/false);
}

__device__ __forceinline__ float block_reduce_sum_256(float v, float* red) {
  const int t = threadIdx.x;
  red[t] = v;
  __syncthreads();
  for (int s = 128; s > 0; s >>= 1) {
    if (t < s) red[t] += red[t + s];
    __syncthreads();
  }
  float r = red[0];
  __syncthreads();
  return r;
}

// ---------------------------------------------------------------------------
// Generic fp32 WMMA GEMM: C = act(A(MxK) * B(KxN) + bias + resid)
// 128x64 tile / block, 8 waves, each wave -> 32x32 (4 16x16 accumulators).
// Double-buffered LDS; all fragments are contiguous 8B-aligned ds_load_b64:
//   As rows padded to 34 floats, B stored transposed BsT[col][k] (pad 34).
// act: 0 = none, 1 = exact GELU.
// ---------------------------------------------------------------------------
#define GBM 128
#define GBN 64
#define GBK 32
#define AF2 ((GBM * GBK / 2) / 256)   // v2f per thread for the A slab = 8
#define BF2 ((GBN * GBK / 2) / 256)   // v2f per thread for the B slab = 4

__global__ void __launch_bounds__(256)
gemm_f32_wmma(const float* __restrict__ A, const float* __restrict__ Bm,
              const float* __restrict__ bias, const float* __restrict__ resid,
              float* __restrict__ C, int M, int N, int K, int act) {
  __shared__ float As[2][GBM][GBK + 2];    // 2*128*34*4 = 34816 B
  __shared__ float BsT[2][GBN][GBK + 2];   // 2* 64*34*4 = 17408 B

  const int bx = blockIdx.x;           // N tiles
  const int by = blockIdx.y;           // M tiles
  const int tid = threadIdx.x;
  const int wave = tid >> 5;
  const int lane = tid & 31;
  const int lo = lane & 15;
  const int hi = lane >> 4;
  const int wr = (wave & 3) * 32;      // wave row base (4 groups of 32)
  const int wc = (wave >> 2) * 32;     // wave col base (2 groups of 32)

  v8f acc00 = {}, acc01 = {}, acc10 = {}, acc11 = {};

  const int nslab = K / GBK;
  v2f ar[AF2];
  v2f br[BF2];

  // ---- stage helpers -------------------------------------------------------
  auto load_slab = [&](int s) {
    const int k0 = s * GBK;
    #pragma unroll
    for (int j = 0; j < AF2; ++j) {
      int e2 = tid + j * 256;
      int r = e2 >> 4, c2 = e2 & 15;   // 16 v2f per 32-float row
      ar[j] = *(const v2f*)&A[(size_t)(by * GBM + r) * K + k0 + c2 * 2];
    }
    #pragma unroll
    for (int j = 0; j < BF2; ++j) {
      int e2 = tid + j * 256;
      int r = e2 >> 5, c2 = e2 & 31;   // 32 v2f per 64-float k-row
      br[j] = *(const v2f*)&Bm[(size_t)(k0 + r) * N + bx * GBN + c2 * 2];
    }
  };
  auto store_slab = [&](int buf) {
    #pragma unroll
    for (int j = 0; j < AF2; ++j) {
      int e2 = tid + j * 256;
      int r = e2 >> 4, c2 = e2 & 15;
      *(v2f*)&As[buf][r][c2 * 2] = ar[j];          // 8B-aligned ds_store_b64
    }
    #pragma unroll
    for (int j = 0; j < BF2; ++j) {
      int e2 = tid + j * 256;
      int r = e2 >> 5, c2 = e2 & 31;
      BsT[buf][c2 * 2 + 0][r] = br[j].x;           // transpose into [col][k]
      BsT[buf][c2 * 2 + 1][r] = br[j].y;
    }
  };

  load_slab(0);
  store_slab(0);
  __syncthreads();

  for (int s = 0; s < nslab; ++s) {
    const int buf = s & 1;
    if (s + 1 < nslab) load_slab(s + 1);   // global loads overlap compute

    #pragma unroll
    for (int kk = 0; kk < GBK; kk += 4) {
      const int kc = kk + 2 * hi;
      v2f a0 = *(const v2f*)&As[buf][wr + lo][kc];
      v2f a1 = *(const v2f*)&As[buf][wr + 16 + lo][kc];
      v2f b0 = *(const v2f*)&BsT[buf][wc + lo][kc];
      v2f b1 = *(const v2f*)&BsT[buf][wc + 16 + lo][kc];
      acc00 = wmma4(a0, b0, acc00);
      acc01 = wmma4(a0, b1, acc01);
      acc10 = wmma4(a1, b0, acc10);
      acc11 = wmma4(a1, b1, acc11);
    }
    if (s + 1 < nslab) {
      store_slab(buf ^ 1);
      __syncthreads();
    }
  }

  // ---- epilogue ------------------------------------------------------------
  const int col0 = bx * GBN + wc + lo;
  const int col1 = col0 + 16;
  #pragma unroll
  for (int i = 0; i < 8; ++i) {
    int r0 = by * GBM + wr + i + 8 * hi;
    int r1 = r0 + 16;
    float v00 = acc00[i], v01 = acc01[i], v10 = acc10[i], v11 = acc11[i];
    if (bias) {
      float bc0 = bias[col0], bc1 = bias[col1];
      v00 += bc0; v01 += bc1; v10 += bc0; v11 += bc1;
    }
    if (resid) {
      v00 += resid[(size_t)r0 * N + col0];
      v01 += resid[(size_t)r0 * N + col1];
      v10 += resid[(size_t)r1 * N + col0];
      v11 += resid[(size_t)r1 * N + col1];
    }
    if (act == 1) {  // exact GELU: 0.5*x*(1+erf(x/sqrt(2)))
      v00 = 0.5f * v00 * (1.0f + erff(v00 * 0.7071067811865476f));
      v01 = 0.5f * v01 * (1.0f + erff(v01 * 0.7071067811865476f));
      v10 = 0.5f * v10 * (1.0f + erff(v10 * 0.7071067811865476f));
      v11 = 0.5f * v11 * (1.0f + erff(v11 * 0.7071067811865476f));
    }
    C[(size_t)r0 * N + col0] = v00;
    C[(size_t)r0 * N + col1] = v01;
    C[(size_t)r1 * N + col0] = v10;
    C[(size_t)r1 * N + col1] = v11;
  }
}

// ---------------------------------------------------------------------------
// q/k transform: scale by 1/sqrt(D), L2-normalize over D, softplus + 1e-6.
// One block (256 threads) per (b,t) row.
// ---------------------------------------------------------------------------
__global__ void __launch_bounds__(256)
qk_transform(const float* __restrict__ qkv, float* __restrict__ qn,
             float* __restrict__ kn) {
  __shared__ float red[256];
  const int row = blockIdx.x;
  const int d = threadIdx.x;
  const float inv_scale = 1.0f / 16.0f;  // 1/sqrt(256)

  float q = qkv[(size_t)row * D3 + d] * inv_scale;
  float k = qkv[(size_t)row * D3 + DD + d] * inv_scale;

  float nq = sqrtf(block_reduce_sum_256(q * q, red));
  float nk = sqrtf(block_reduce_sum_256(k * k, red));
  q = q / fmaxf(nq, 1e-12f);
  k = k / fmaxf(nk, 1e-12f);
  // numerically stable softplus
  q = fmaxf(q, 0.0f) + log1pf(expf(-fabsf(q))) + 1e-6f;
  k = fmaxf(k, 0.0f) + log1pf(expf(-fabsf(k))) + 1e-6f;
  qn[(size_t)row * DD + d] = q;
  kn[(size_t)row * DD + d] = k;
}

// ---------------------------------------------------------------------------
// Cumulative sum of k over time, per (b, d) lane.  Grid = B, block = 256.
// ---------------------------------------------------------------------------
__global__ void __launch_bounds__(256)
ksum_kernel(const float* __restrict__ kn, float* __restrict__ ksum) {
  const int b = blockIdx.x;
  const int d = threadIdx.x;
  float run = 0.0f;
  for (int t = 0; t < TT; ++t) {
    size_t idx = ((size_t)(b * TT + t)) * DD + d;
    run += kn[idx];
    ksum[idx] = run;
  }
}

// ---------------------------------------------------------------------------
// Chunked causal linear-attention scan, WMMA version.
// Grid = B * (D/16) single-wave workgroups.  Each wave owns a 256x16 slice of
// the running state S (k-dim x e-dim), stored TRANSPOSED in LDS as St[e][d] so
// B-fragments are contiguous ds_load_b64.  Walks T in chunks of 16:
//   num = Q @ S_prev + (Q Kᵀ ∘ causal) @ V ;  S += Kᵀ @ V
// ---------------------------------------------------------------------------
__global__ void __launch_bounds__(32)
attn_scan(const float* __restrict__ qn, const float* __restrict__ kn,
          const float* __restrict__ qkv /* for v columns */,
          float* __restrict__ num) {
  __shared__ float St[16][258];    // state slice, [e][d], pad -> 8B-aligned rows
  __shared__ float Qs[16][258];    // chunk Q [t][d]
  __shared__ float Ks[16][258];    // chunk K [t][d]
  __shared__ float Aqk[16][18];    // masked Q Kᵀ tile [tq][tk]

  const int b = blockIdx.x >> 4;
  const int eb = blockIdx.x & 15;          // 16-wide e slice
  const int lane = threadIdx.x;
  const int lo = lane & 15;
  const int hi = lane >> 4;

  for (int i = lane; i < 16 * DD; i += 32) St[i >> 8][i & 255] = 0.0f;
  __syncthreads();

  const float* qb = qn + (size_t)b * TT * DD;
  const float* kb = kn + (size_t)b * TT * DD;
  const float* vb = qkv + (size_t)b * TT * D3 + 2 * DD + eb * 16;
  float* nb = num + (size_t)b * TT * DD + eb * 16;

  for (int c = 0; c < TT / 16; ++c) {
    const int t0 = c * 16;
    // stage Q,K chunk (coalesced 32-float runs)
    for (int i = lane; i < 16 * DD; i += 32) {
      int r = i >> 8, cc = i & 255;
      Qs[r][cc] = qb[(size_t)(t0 + r) * DD + cc];
      Ks[r][cc] = kb[(size_t)(t0 + r) * DD + cc];
    }
    __syncthreads();

    // V tile B-fragments for the 4 k-steps of the 16-deep contraction
    v2f vf[4];
    #pragma unroll
    for (int kk = 0; kk < 4; ++kk) {
      vf[kk].x = vb[(size_t)(t0 + kk * 4 + 2 * hi) * D3 + lo];
      vf[kk].y = vb[(size_t)(t0 + kk * 4 + 2 * hi + 1) * D3 + lo];
    }

    // inter-chunk: num += Q(16x256) @ S(256x16); b-frag = St[lo][k..k+1]
    v8f accN = {};
    #pragma unroll 8
    for (int k = 0; k < DD; k += 4) {
      v2f a  = *(const v2f*)&Qs[lo][k + 2 * hi];
      v2f bf = *(const v2f*)&St[lo][k + 2 * hi];
      accN = wmma4(a, bf, accN);
    }

    // intra-chunk attention matrix A = Q Kᵀ (B operand = Kᵀ: contiguous in Ks)
    v8f accA = {};
    #pragma unroll 8
    for (int k = 0; k < DD; k += 4) {
      v2f a  = *(const v2f*)&Qs[lo][k + 2 * hi];
      v2f bf = *(const v2f*)&Ks[lo][k + 2 * hi];
      accA = wmma4(a, bf, accA);
    }
    // causal mask (keep key index <= query index, incl. diagonal) -> LDS
    #pragma unroll
    for (int i = 0; i < 8; ++i) {
      int r = i + 8 * hi;                       // query t
      Aqk[r][lo] = (lo <= r) ? accA[i] : 0.0f;  // lo = key t
    }
    __syncthreads();

    // num += (A ∘ mask)(16x16) @ V(16x16)
    #pragma unroll
    for (int kk = 0; kk < 4; ++kk) {
      v2f a = *(const v2f*)&Aqk[lo][kk * 4 + 2 * hi];
      accN = wmma4(a, vf[kk], accN);
    }

    // write numerator tile
    #pragma unroll
    for (int i = 0; i < 8; ++i)
      nb[(size_t)(t0 + i + 8 * hi) * DD + lo] = accN[i];

    // state update: S(256x16) += Kᵀ(256x16) @ V(16x16), 16 d-tiles
    for (int dt = 0; dt < 16; ++dt) {
      v8f cs;
      #pragma unroll
      for (int i = 0; i < 8; ++i) cs[i] = St[lo][dt * 16 + i + 8 * hi];
      #pragma unroll
      for (int kk = 0; kk < 4; ++kk) {
        v2f a;  // A = Kᵀ: A[d][t] = Ks[t][d]
        a.x = Ks[kk * 4 + 2 * hi][dt * 16 + lo];
        a.y = Ks[kk * 4 + 2 * hi + 1][dt * 16 + lo];
        cs = wmma4(a, vf[kk], cs);
      }
      #pragma unroll
      for (int i = 0; i < 8; ++i) St[lo][dt * 16 + i + 8 * hi] = cs[i];
    }
    __syncthreads();
  }
}

// ---------------------------------------------------------------------------
// attn_out = clip(num, ±1e4) / max(q · ksum, 1e-6).  One block per row.
// ---------------------------------------------------------------------------
__global__ void __launch_bounds__(256)
attn_finalize(const float* __restrict__ qn, const float* __restrict__ ksum,
              const float* __restrict__ num, float* __restrict__ attn) {
  __shared__ float red[256];
  const int row = blockIdx.x;
  const int d = threadIdx.x;
  size_t idx = (size_t)row * DD + d;
  float den = block_reduce_sum_256(qn[idx] * ksum[idx], red);
  den = fmaxf(den, 1e-6f);
  float nv = num[idx];
  nv = fminf(fmaxf(nv, -10000.0f), 10000.0f);
  attn[idx] = nv / den;
}

// ---------------------------------------------------------------------------
// LayerNorm over D=256. One block per row.
// ---------------------------------------------------------------------------
__global__ void __launch_bounds__(256)
layernorm_k(const float* __restrict__ x, const float* __restrict__ g,
            const float* __restrict__ bta, float* __restrict__ out) {
  __shared__ float red[256];
  const int row = blockIdx.x;
  const int d = threadIdx.x;
  size_t idx = (size_t)row * DD + d;
  float v = x[idx];
  float mu = block_reduce_sum_256(v, red) * (1.0f / DD);
  float dv = v - mu;
  float var = block_reduce_sum_256(dv * dv, red) * (1.0f / DD);
  out[idx] = dv * rsqrtf(var + 1e-5f) * g[d] + bta[d];
}

// ---------------------------------------------------------------------------
extern "C" void kernel_launch(void* const* d_in, const int* in_sizes, int n_in,
                              void* d_out, int out_size, void* d_ws, size_t ws_size,
                              hipStream_t stream) {
  const float* x      = (const float*)d_in[0];
  const float* qkv_w  = (const float*)d_in[1];
  const float* qkv_b  = (const float*)d_in[2];
  const float* proj_w = (const float*)d_in[3];
  const float* proj_b = (const float*)d_in[4];
  const float* ln1_g  = (const float*)d_in[5];
  const float* ln1_b  = (const float*)d_in[6];
  const float* w1     = (const float*)d_in[7];
  const float* b1     = (const float*)d_in[8];
  const float* w2     = (const float*)d_in[9];
  const float* b2     = (const float*)d_in[10];
  const float* ln2_g  = (const float*)d_in[11];
  const float* ln2_b  = (const float*)d_in[12];
  float* out = (float*)d_out;

  float* ws    = (float*)d_ws;
  float* qkv   = ws;                       // BT x 768
  float* qn    = qkv  + (size_t)BT * D3;   // BT x 256
  float* kn    = qn   + (size_t)BT * DD;
  float* ksum  = kn   + (size_t)BT * DD;
  float* num   = ksum + (size_t)BT * DD;
  float* attn  = num  + (size_t)BT * DD;
  float* y1    = attn + (size_t)BT * DD;
  float* xattn = y1   + (size_t)BT * DD;
  float* h     = xattn + (size_t)BT * DD;  // BT x 1024
  float* y2    = h    + (size_t)BT * HH;

  dim3 blk(256);

  // 1) qkv = x @ qkv_w + qkv_b
  gemm_f32_wmma<<<dim3(D3 / GBN, BT / GBM), blk, 0, stream>>>(
      x, qkv_w, qkv_b, nullptr, qkv, BT, D3, DD, 0);

  // 2) q/k normalize + softplus
  qk_transform<<<BT, blk, 0, stream>>>(qkv, qn, kn);

  // 3) cumulative key sums (for denominator)
  ksum_kernel<<<BB, blk, 0, stream>>>(kn, ksum);

  // 4) chunked causal linear-attention scan (numerator)
  attn_scan<<<BB * (DD / 16), dim3(32), 0, stream>>>(qn, kn, qkv, num);

  // 5) attn_out = clip(num)/max(q·ksum, 1e-6)
  attn_finalize<<<BT, blk, 0, stream>>>(qn, ksum, num, attn);

  // 6) y1 = x + attn @ proj_w + proj_b
  gemm_f32_wmma<<<dim3(DD / GBN, BT / GBM), blk, 0, stream>>>(
      attn, proj_w, proj_b, x, y1, BT, DD, DD, 0);

  // 7) x_attn = LN1(y1)
  layernorm_k<<<BT, blk, 0, stream>>>(y1, ln1_g, ln1_b, xattn);

  // 8) h = gelu(x_attn @ w1 + b1)
  gemm_f32_wmma<<<dim3(HH / GBN, BT / GBM), blk, 0, stream>>>(
      xattn, w1, b1, nullptr, h, BT, HH, DD, 1);

  // 9) y2 = x_attn + h @ w2 + b2
  gemm_f32_wmma<<<dim3(DD / GBN, BT / GBM), blk, 0, stream>>>(
      h, w2, b2, xattn, y2, BT, DD, HH, 0);

  // 10) out = LN2(y2)
  layernorm_k<<<BT, blk, 0, stream>>>(y2, ln2_g, ln2_b, out);
}